// InferenceModule_42245298323617
// MI455X (gfx1250) — compile-verified
//
#include <hip/hip_runtime.h>

#define Bb 64
#define Nn 4
#define Ss 512
#define Ee 256
#define Rr 64
#define Vv 32000
#define EPSM 1e-6f
#define LNEPS 1e-6f

typedef __attribute__((ext_vector_type(2))) float v2f;
typedef __attribute__((ext_vector_type(8))) float v8f;

// ---------------------------------------------------------------------------
// Kernel 1: slot selection (near-one-hot mix over N=4) + the 4 projections.
//   sel_k[b,s] = eps*sum_n slots[b,n,s] + (1-4eps)*slots[b,k,s]
//   e1 = sel_0 @ e1_w^T + e1_b ; r_k = sel_k @ r_kw^T + r_kb
// One block per batch. sel staged in LDS (8 KB).
// ---------------------------------------------------------------------------
__global__ __launch_bounds__(256) void k_proj(
    const float* __restrict__ slots,
    const float* __restrict__ e1_w, const float* __restrict__ e1_b,
    const float* __restrict__ r1_w, const float* __restrict__ r1_b,
    const float* __restrict__ r2_w, const float* __restrict__ r2_b,
    const float* __restrict__ r3_w, const float* __restrict__ r3_b,
    float* __restrict__ e1_out, float* __restrict__ r_out) {
  __shared__ float sel[4][Ss];
  const int b = blockIdx.x;
  const int tid = threadIdx.x;
  const float* sb = slots + (size_t)b * Nn * Ss;
  const float w = 1.0f - 4.0f * EPSM;
  for (int s = tid; s < Ss; s += 256) {
    float s0 = sb[0 * Ss + s], s1 = sb[1 * Ss + s];
    float s2 = sb[2 * Ss + s], s3 = sb[3 * Ss + s];
    float tot = s0 + s1 + s2 + s3;
    sel[0][s] = EPSM * tot + w * s0;
    sel[1][s] = EPSM * tot + w * s1;
    sel[2][s] = EPSM * tot + w * s2;
    sel[3][s] = EPSM * tot + w * s3;
  }
  __syncthreads();
  {  // e1: 256 threads, one output each, K=512
    const int j = tid;
    const float* wr = e1_w + (size_t)j * Ss;
    float acc = e1_b[j];
    #pragma unroll 4
    for (int s = 0; s < Ss; ++s) acc = fmaf(sel[0][s], wr[s], acc);
    e1_out[b * Ee + j] = acc;
  }
  if (tid < 192) {  // r1/r2/r3: 3*64 outputs
    const int k = tid >> 6;
    const int j = tid & 63;
    const float* wr;
    const float* bb;
    if (k == 0)      { wr = r1_w; bb = r1_b; }
    else if (k == 1) { wr = r2_w; bb = r2_b; }
    else             { wr = r3_w; bb = r3_b; }
    wr += (size_t)j * Ss;
    const float* sl = sel[k + 1];
    float acc = bb[j];
    #pragma unroll 4
    for (int s = 0; s < Ss; ++s) acc = fmaf(sl[s], wr[s], acc);
    r_out[(k * Bb + b) * Rr + j] = acc;
  }
}

// ---------------------------------------------------------------------------
// Kernel 2: THE streaming pass over TPR (1.07 GB, read exactly once).
//   Wk[b,e,f] = sum_r rk[b,r] * TPR[b,e,r,f]   for k=0..2 simultaneously.
// Grid = B*E blocks; thread f reads TPR[b,e,r,f] coalesced across f.
// 3 FMAs per 4 bytes loaded -> pure HBM bandwidth.
// ---------------------------------------------------------------------------
__global__ __launch_bounds__(256) void k_tpr(
    const float* __restrict__ TPR, const float* __restrict__ r_in,
    float* __restrict__ W) {
  __shared__ float rl[3 * Rr];
  const int b = blockIdx.x >> 8;   // E = 256
  const int e = blockIdx.x & 255;
  const int f = threadIdx.x;
  if (f < 3 * Rr) {
    const int k = f >> 6, j = f & 63;
    rl[f] = r_in[(k * Bb + b) * Rr + j];
  }
  __syncthreads();
  const float* tp = TPR + ((size_t)(b * Ee + e) * Rr) * Ee + f;
  float a1 = 0.f, a2 = 0.f, a3 = 0.f;
  #pragma unroll 8
  for (int r = 0; r < Rr; ++r) {
    const float t = tp[(size_t)r * Ee];
    a1 = fmaf(rl[r], t, a1);
    a2 = fmaf(rl[Rr + r], t, a2);
    a3 = fmaf(rl[2 * Rr + r], t, a3);
  }
  const size_t KS = (size_t)Bb * Ee * Ee;
  const size_t idx = ((size_t)b * Ee + e) * Ee + f;
  W[idx]          = a1;
  W[KS + idx]     = a2;
  W[2 * KS + idx] = a3;
}

// ---------------------------------------------------------------------------
// Kernel 3: sequential chain  i_{k} = LN(evec @ Wk[b])  with block reductions.
// One block per batch; evec lives in LDS between steps.
// ---------------------------------------------------------------------------
__global__ __launch_bounds__(256) void k_chain(
    const float* __restrict__ e1_in, const float* __restrict__ W,
    const float* __restrict__ ln1_g, const float* __restrict__ ln1_b,
    const float* __restrict__ ln2_g, const float* __restrict__ ln2_b,
    const float* __restrict__ ln3_g, const float* __restrict__ ln3_b,
    float* __restrict__ ss_out) {
  __shared__ float evec[Ee];
  __shared__ float red[Ee];
  const int b = blockIdx.x;
  const int f = threadIdx.x;
  evec[f] = e1_in[b * Ee + f];
  float ssum = 0.f;
  __syncthreads();
  const size_t KS = (size_t)Bb * Ee * Ee;
  for (int k = 0; k < 3; ++k) {
    const float* wk = W + (size_t)k * KS + ((size_t)b * Ee) * Ee + f;
    float acc = 0.f;
    #pragma unroll 4
    for (int e = 0; e < Ee; ++e) acc = fmaf(evec[e], wk[(size_t)e * Ee], acc);
    __syncthreads();
    // mean
    red[f] = acc;
    __syncthreads();
    for (int off = 128; off > 0; off >>= 1) {
      if (f < off) red[f] += red[f + off];
      __syncthreads();
    }
    const float mean = red[0] * (1.0f / Ee);
    __syncthreads();
    // variance
    const float d = acc - mean;
    red[f] = d * d;
    __syncthreads();
    for (int off = 128; off > 0; off >>= 1) {
      if (f < off) red[f] += red[f + off];
      __syncthreads();
    }
    const float var = red[0] * (1.0f / Ee);
    __syncthreads();
    const float* g  = (k == 0) ? ln1_g : ((k == 1) ? ln2_g : ln3_g);
    const float* be = (k == 0) ? ln1_b : ((k == 1) ? ln2_b : ln3_b);
    const float val = d * rsqrtf(var + LNEPS) * g[f] + be[f];
    ssum += val;
    evec[f] = val;
    __syncthreads();
  }
  ss_out[b * Ee + f] = ssum;
}

// ---------------------------------------------------------------------------
// Kernel 4: logits = step_sum @ Z_w^T + Z_b using V_WMMA_F32_16X16X4_F32.
// M=64 (B), N=32000 (V), K=256 (E). One wave per 16x16 tile; 4 waves/block
// cover all 4 M-tiles of one 16-wide V stripe. Fragments per ISA 32-bit
// layouts: lanes 0-15 carry K={k0,k0+1}, lanes 16-31 K={k0+2,k0+3} -> both
// A and B fragments are aligned float2 loads streaming along K.
// ---------------------------------------------------------------------------
__global__ __launch_bounds__(128) void k_logits(
    const float* __restrict__ ss, const float* __restrict__ Zw,
    const float* __restrict__ Zb, float* __restrict__ out) {
  const int v0   = blockIdx.x * 16;
  const int wave = threadIdx.x >> 5;
  const int lane = threadIdx.x & 31;
  const int m0   = wave * 16;
  const int mn   = lane & 15;            // A row / B col within the tile
  const int kh   = (lane >> 4) * 2;      // K half-select
  const float* aptr = ss + (size_t)(m0 + mn) * Ee + kh;
  const float* bptr = Zw + (size_t)(v0 + mn) * Ee + kh;  // B[k,n] = Zw[v0+n,k]
  v8f c = {};
  for (int k0 = 0; k0 < Ee; k0 += 4) {
    v2f a = *(const v2f*)(aptr + k0);
    v2f bfrag = *(const v2f*)(bptr + k0);
    c = __builtin_amdgcn_wmma_f32_16x16x4_f32(
        /*neg_a=*/false, a, /*neg_b=*/false, bfrag,
        /*c_mod=*/(short)0, c, /*reuse_a=*/false, /*reuse_b=*/false);
  }
  // D layout: VGPR i -> M = i + 8*(lane>>4), N = lane&15
  const int n  = lane & 15;
  const int mh = (lane >> 4) * 8;
  const float bias = Zb[v0 + n];
  #pragma unroll
  for (int i = 0; i < 8; ++i) {
    out[(size_t)(m0 + mh + i) * Vv + (v0 + n)] = c[i] + bias;
  }
}

// ---------------------------------------------------------------------------
extern "C" void kernel_launch(void* const* d_in, const int* in_sizes, int n_in,
                              void* d_out, int out_size, void* d_ws,
                              size_t ws_size, hipStream_t stream) {
  (void)in_sizes; (void)n_in; (void)out_size; (void)ws_size;
  const float* slots = (const float*)d_in[0];
  const float* TPR   = (const float*)d_in[1];
  const float* e1_w  = (const float*)d_in[2];
  const float* e1_b  = (const float*)d_in[3];
  const float* r1_w  = (const float*)d_in[4];
  const float* r1_b  = (const float*)d_in[5];
  const float* r2_w  = (const float*)d_in[6];
  const float* r2_b  = (const float*)d_in[7];
  const float* r3_w  = (const float*)d_in[8];
  const float* r3_b  = (const float*)d_in[9];
  const float* ln1_g = (const float*)d_in[10];
  const float* ln1_b = (const float*)d_in[11];
  const float* ln2_g = (const float*)d_in[12];
  const float* ln2_b = (const float*)d_in[13];
  const float* ln3_g = (const float*)d_in[14];
  const float* ln3_b = (const float*)d_in[15];
  const float* Z_w   = (const float*)d_in[16];
  const float* Z_b   = (const float*)d_in[17];

  float* W  = (float*)d_ws;                       // [3, B, E, E]  ~50.3 MB
  float* e1 = W + (size_t)3 * Bb * Ee * Ee;       // [B, E]
  float* rv = e1 + (size_t)Bb * Ee;               // [3, B, R]
  float* ss = rv + (size_t)3 * Bb * Rr;           // [B, E]

  k_proj<<<Bb, 256, 0, stream>>>(slots, e1_w, e1_b, r1_w, r1_b, r2_w, r2_b,
                                 r3_w, r3_b, e1, rv);
  k_tpr<<<Bb * Ee, 256, 0, stream>>>(TPR, rv, W);
  k_chain<<<Bb, 256, 0, stream>>>(e1, W, ln1_g, ln1_b, ln2_g, ln2_b, ln3_g,
                                  ln3_b, ss);
  k_logits<<<Vv / 16, 128, 0, stream>>>(ss, Z_w, Z_b, (float*)d_out);
}